// Decoder_49409303773330
// MI455X (gfx1250) — compile-verified
//
#include <hip/hip_runtime.h>
#include <math.h>

#define B_   128
#define T_   400
#define H_   256
#define E_   128
#define V_   50000
#define OOV_ 100
#define H2   512          // 2H
#define H4   1024         // 4H
#define KXC  640          // 2H+E
#define KG   384          // E+H
#define KPG  1152         // 4H+E
#define K1K  768          // 3H
#define VE   50100        // V+OOV

typedef __attribute__((ext_vector_type(2))) float v2f;
typedef __attribute__((ext_vector_type(8))) float v8f;

__device__ __forceinline__ float sigm_(float x) { return 1.0f / (1.0f + __expf(-x)); }

// CDNA5 hardware tanh (transcendental VALU op)
__device__ __forceinline__ float fast_tanh(float x) {
  float r;
  asm volatile("v_tanh_f32 %0, %1" : "=v"(r) : "v"(x));
  return r;
}

// =====================================================================
// Generic f32 WMMA GEMM:  D[128, N] = A[128, K] @ W[N, K]^T + bias[N]
// K and LDD are compile-time so all A-fragment / D-store offsets lower to
// immediate-offset vmem/ds instructions (no dependent 64-bit address adds
// in the hot loop). One wave owns a 16-wide N tile and loops all 8 M tiles
// (M = 128), reusing the B fragment 8x; fragment loads for a k-step are
// batched ahead of the 8 WMMAs so waits stagger instead of serializing.
// STAGE=1: A staged in dynamic LDS per block (large-N case: A reuse kept
// on-WGP instead of hammering L2).
// Layouts per CDNA5 ISA for V_WMMA_F32_16X16X4_F32:
//   A 16x4 : lanes 0-15 hold K+0/K+1 (M=lane), lanes 16-31 hold K+2/K+3
//   B 4x16 : mirrored (N=lane&15)
//   C/D    : vgpr r -> M=r (lanes 0-15) / M=r+8 (lanes 16-31), N=lane&15
// =====================================================================
template <int STAGE, int K, int LDD>
__global__ __launch_bounds__(256) void k_gemm_wmma(
    const float* __restrict__ A, const float* __restrict__ W,
    const float* __restrict__ bias, float* __restrict__ D, int N) {
  extern __shared__ float sA[];
  if (STAGE) {
    const float4* src = (const float4*)A;
    float4* dst = (float4*)sA;
    constexpr int n4 = (B_ * K) >> 2;
    for (int i = threadIdx.x; i < n4; i += 256) dst[i] = src[i];
    __syncthreads();
  }
  int wave = (blockIdx.x * blockDim.x + threadIdx.x) >> 5;
  int lane = threadIdx.x & 31;
  int col0 = wave * 16;
  if (col0 >= N) return;                      // wave-uniform: EXEC stays all-1s for active waves
  int l15 = lane & 15;
  int khi = (lane >> 4) << 1;                 // lanes 0-15: K+0/+1, lanes 16-31: K+2/+3

  v8f acc[8];
  #pragma unroll
  for (int m = 0; m < 8; ++m)
    #pragma unroll
    for (int r = 0; r < 8; ++r) acc[m][r] = 0.f;

  const float* bb = W + (size_t)(col0 + l15) * K + khi;

  #pragma unroll 2
  for (int kk = 0; kk < K; kk += 4) {
    v2f bm = *(const v2f*)(bb + kk);
    v2f am[8];
    if (STAGE) {
      const float* ab = sA + l15 * K + khi + kk;
      #pragma unroll
      for (int mt = 0; mt < 8; ++mt) am[mt] = *(const v2f*)(ab + mt * 16 * K);
    } else {
      const float* ab = A + (size_t)l15 * K + khi + kk;
      #pragma unroll
      for (int mt = 0; mt < 8; ++mt) am[mt] = *(const v2f*)(ab + mt * 16 * K);
    }
    #pragma unroll
    for (int mt = 0; mt < 8; ++mt)
      acc[mt] = __builtin_amdgcn_wmma_f32_16x16x4_f32(
          false, am[mt], false, bm, (short)0, acc[mt], false, false);
  }

  int n = col0 + l15;
  float bv = bias ? bias[n] : 0.f;
  float* drow = D + n;
  #pragma unroll
  for (int mt = 0; mt < 8; ++mt) {
    int rowbase = mt * 16 + ((lane >> 4) << 3);
    #pragma unroll
    for (int r = 0; r < 8; ++r)
      drow[(size_t)(rowbase + r) * LDD] = acc[mt][r] + bv;
  }
}

// ---------------- packing / concat helpers (tiny copies) ----------------
__global__ void k_cat_xc(const int* __restrict__ y, const float* __restrict__ ctm1,
                         const float* __restrict__ glove, float* __restrict__ Axc) {
  int i = blockIdx.x * blockDim.x + threadIdx.x;              // B*KXC
  if (i >= B_ * KXC) return;
  int b = i / KXC, k = i % KXC;
  Axc[i] = (k < H2) ? ctm1[(size_t)b * H2 + k]
                    : glove[(size_t)y[b] * E_ + (k - H2)];
}

__global__ void k_cat_g(const float* __restrict__ x, const float* __restrict__ h0,
                        float* __restrict__ Ag) {
  int i = blockIdx.x * blockDim.x + threadIdx.x;              // B*KG
  if (i >= B_ * KG) return;
  int b = i / KG, k = i % KG;
  Ag[i] = (k < E_) ? x[(size_t)b * E_ + k] : h0[(size_t)b * H_ + (k - E_)];
}

__global__ void k_pack_wg(const float* __restrict__ Wih, const float* __restrict__ Whh,
                          float* __restrict__ Wg) {
  int i = blockIdx.x * blockDim.x + threadIdx.x;              // H4*KG
  if (i >= H4 * KG) return;
  int j = i / KG, k = i % KG;
  Wg[i] = (k < E_) ? Wih[(size_t)j * E_ + k] : Whh[(size_t)j * H_ + (k - E_)];
}

__global__ void k_cat_dp(const float* __restrict__ h_new, const float* __restrict__ c_new,
                         float* __restrict__ Adp) {
  int i = blockIdx.x * blockDim.x + threadIdx.x;              // B*H2 (this is s_t_hat)
  if (i >= B_ * H2) return;
  int b = i / H2, k = i % H2;
  Adp[i] = (k < H_) ? h_new[(size_t)b * H_ + k] : c_new[(size_t)b * H_ + (k - H_)];
}

__global__ void k_cat_o1(const float* __restrict__ h_new, const float* __restrict__ c_t,
                         float* __restrict__ Ao1) {
  int i = blockIdx.x * blockDim.x + threadIdx.x;              // B*K1K
  if (i >= B_ * K1K) return;
  int b = i / K1K, k = i % K1K;
  Ao1[i] = (k < H_) ? h_new[(size_t)b * H_ + k] : c_t[(size_t)b * H2 + (k - H_)];
}

// ---------------- LSTM elementwise (adds both biases here) ----------------
__global__ void k_lstm(const float* __restrict__ gates, const float* __restrict__ cprev,
                       const float* __restrict__ bih, const float* __restrict__ bhh,
                       float* __restrict__ h_new, float* __restrict__ c_new) {
  int i = blockIdx.x * blockDim.x + threadIdx.x;              // B*H
  if (i >= B_ * H_) return;
  int b = i / H_, h = i % H_;
  const float* g = gates + (size_t)b * H4;
  float ig = sigm_(g[h]            + bih[h]            + bhh[h]);
  float fg = sigm_(g[H_ + h]       + bih[H_ + h]       + bhh[H_ + h]);
  float gg = fast_tanh(g[2*H_ + h] + bih[2*H_ + h]     + bhh[2*H_ + h]);
  float og = sigm_(g[3*H_ + h]     + bih[3*H_ + h]     + bhh[3*H_ + h]);
  float c = fg * cprev[i] + ig * gg;
  c_new[i] = c;
  h_new[i] = og * fast_tanh(c);
}

// ---------------- attention scores (streams 105 MB, b128 loads + v_tanh) ----------------
__global__ void k_scores(const float* __restrict__ ef, const float* __restrict__ dec,
                         const float* __restrict__ cov, const float* __restrict__ vatt,
                         const float* __restrict__ wc, float* __restrict__ scores) {
  __shared__ float sv[H2], sw[H2], sd[H2];
  int b  = blockIdx.x / (T_ / 8);
  int tg = blockIdx.x % (T_ / 8);
  for (int i = threadIdx.x; i < H2; i += blockDim.x) {
    sv[i] = vatt[i]; sw[i] = wc[i]; sd[i] = dec[(size_t)b * H2 + i];
  }
  __syncthreads();
  int wave = threadIdx.x >> 5, lane = threadIdx.x & 31;
  int t = tg * 8 + wave;
  const float4* er = (const float4*)(ef + ((size_t)b * T_ + t) * H2);
  float cv = cov[b * T_ + t];
  float acc = 0.f;
  #pragma unroll
  for (int j = 0; j < 4; ++j) {
    int idx = j * 32 + lane;
    float4 ev = er[idx];
    int d = idx * 4;
    acc += sv[d + 0] * fast_tanh(ev.x + sd[d + 0] + cv * sw[d + 0]);
    acc += sv[d + 1] * fast_tanh(ev.y + sd[d + 1] + cv * sw[d + 1]);
    acc += sv[d + 2] * fast_tanh(ev.z + sd[d + 2] + cv * sw[d + 2]);
    acc += sv[d + 3] * fast_tanh(ev.w + sd[d + 3] + cv * sw[d + 3]);
  }
  for (int off = 16; off; off >>= 1) acc += __shfl_down(acc, off, 32);
  if (lane == 0) scores[b * T_ + t] = acc;
}

// ---------------- softmax over T + mask + renorm + coverage_next ----------------
__global__ void k_attn(const float* __restrict__ scores, const float* __restrict__ mask,
                       const float* __restrict__ cov, float* __restrict__ attn,
                       float* __restrict__ cov_next) {
  __shared__ float red[512];
  int b = blockIdx.x, t = threadIdx.x;
  float s = (t < T_) ? scores[b * T_ + t] : -1e30f;
  red[t] = s; __syncthreads();
  for (int off = 256; off; off >>= 1) { if (t < off) red[t] = fmaxf(red[t], red[t + off]); __syncthreads(); }
  float mx = red[0]; __syncthreads();
  float e = (t < T_) ? __expf(s - mx) : 0.f;
  red[t] = e; __syncthreads();
  for (int off = 256; off; off >>= 1) { if (t < off) red[t] += red[t + off]; __syncthreads(); }
  float denom = red[0]; __syncthreads();
  float a = (t < T_) ? (e / denom) * mask[b * T_ + t] : 0.f;
  red[t] = a; __syncthreads();
  for (int off = 256; off; off >>= 1) { if (t < off) red[t] += red[t + off]; __syncthreads(); }
  float asum = red[0];
  if (t < T_) {
    float av = a / asum;
    attn[b * T_ + t] = av;
    cov_next[b * T_ + t] = cov[b * T_ + t] + av;
  }
}

// ---------------- c_t = attn @ encoder_outputs (second 105 MB stream) ----------------
__global__ void k_ct(const float* __restrict__ attn, const float* __restrict__ eo,
                     float* __restrict__ c_t) {
  __shared__ float sa[T_];
  int b = blockIdx.x, d = threadIdx.x;                        // blockDim = 512 = 2H
  for (int i = threadIdx.x; i < T_; i += blockDim.x) sa[i] = attn[b * T_ + i];
  __syncthreads();
  const float* er = eo + (size_t)b * T_ * H2 + d;
  float acc = 0.f;
  for (int t = 0; t < T_; ++t) acc += sa[t] * er[(size_t)t * H2];
  c_t[(size_t)b * H2 + d] = acc;
}

// ---------------- p_gen gate (one wave per b) ----------------
__global__ void k_pgen(const float* __restrict__ c_t, const float* __restrict__ sth,
                       const float* __restrict__ x, const float* __restrict__ Wpg,
                       const float* __restrict__ bpg, float* __restrict__ pg) {
  int b = (blockIdx.x * blockDim.x + threadIdx.x) >> 5;       // 128 waves total
  int lane = threadIdx.x & 31;
  float acc = 0.f;
  for (int k = lane; k < KPG; k += 32) {
    float v;
    if (k < H2)      v = c_t[(size_t)b * H2 + k];
    else if (k < H4) v = sth[(size_t)b * H2 + (k - H2)];
    else             v = x[(size_t)b * E_ + (k - H4)];
    acc += Wpg[k] * v;
  }
  for (int off = 16; off; off >>= 1) acc += __shfl_down(acc, off, 32);
  if (lane == 0) pg[b] = 1.f / (1.f + __expf(-(acc + bpg[0])));
}

// ---------------- online softmax over V (2 passes), scale by p_gen, zero OOV ----------------
__global__ void k_vsoftmax(const float* __restrict__ pg, float* __restrict__ fdist) {
  __shared__ float rm[1024], rs[1024];
  int b = blockIdx.x, tid = threadIdx.x;
  float* row = fdist + (size_t)b * VE;
  float m = -1e30f, s = 0.f;
  for (int v = tid; v < V_; v += 1024) {
    float xv = row[v];
    float nm = fmaxf(m, xv);
    s = s * __expf(m - nm) + __expf(xv - nm);
    m = nm;
  }
  rm[tid] = m; rs[tid] = s; __syncthreads();
  for (int off = 512; off; off >>= 1) {
    if (tid < off) {
      float m2 = rm[tid + off], s2 = rs[tid + off];
      float nm = fmaxf(rm[tid], m2);
      rs[tid] = rs[tid] * __expf(rm[tid] - nm) + s2 * __expf(m2 - nm);
      rm[tid] = nm;
    }
    __syncthreads();
  }
  float M = rm[0], scale = pg[b] / rs[0];
  for (int v = tid; v < V_; v += 1024) row[v] = __expf(row[v] - M) * scale;
  for (int v = V_ + tid; v < VE; v += 1024) row[v] = 0.f;
}

// ---------------- scatter-add (1-p_gen)*attn into extended vocab ----------------
__global__ void k_scatter(const float* __restrict__ attn, const float* __restrict__ pg,
                          const int* __restrict__ ext, float* __restrict__ fdist) {
  int i = blockIdx.x * blockDim.x + threadIdx.x;              // B*T
  if (i >= B_ * T_) return;
  int b = i / T_;
  atomicAdd(fdist + (size_t)b * VE + ext[i], (1.f - pg[b]) * attn[i]);
}

extern "C" void kernel_launch(void* const* d_in, const int* in_sizes, int n_in,
                              void* d_out, int out_size, void* d_ws, size_t ws_size,
                              hipStream_t stream) {
  const int*   y     = (const int*)d_in[0];
  const float* h0    = (const float*)d_in[1];
  const float* c0    = (const float*)d_in[2];
  const float* eo    = (const float*)d_in[3];
  const float* ef    = (const float*)d_in[4];
  const float* mask  = (const float*)d_in[5];
  const float* ctm1  = (const float*)d_in[6];
  const int*   ext   = (const int*)d_in[8];
  const float* cov   = (const float*)d_in[9];
  const float* glove = (const float*)d_in[10];
  const float* Wxc   = (const float*)d_in[11];
  const float* bxc   = (const float*)d_in[12];
  const float* Wih   = (const float*)d_in[13];
  const float* Whh   = (const float*)d_in[14];
  const float* bih   = (const float*)d_in[15];
  const float* bhh   = (const float*)d_in[16];
  const float* Wc    = (const float*)d_in[17];
  const float* Wdp   = (const float*)d_in[18];
  const float* bdp   = (const float*)d_in[19];
  const float* vatt  = (const float*)d_in[20];
  const float* Wpg   = (const float*)d_in[21];
  const float* bpg   = (const float*)d_in[22];
  const float* W1    = (const float*)d_in[23];
  const float* b1    = (const float*)d_in[24];
  const float* W2    = (const float*)d_in[25];
  const float* b2    = (const float*)d_in[26];

  // outputs, concatenated flat in return order
  float* o       = (float*)d_out;
  float* fdist   = o;                                   // [B, VE]
  float* h_new   = o + (size_t)B_ * VE;                 // [1,B,H]
  float* c_new   = h_new + B_ * H_;                     // [1,B,H]
  float* c_t     = c_new + B_ * H_;                     // [B,2H]
  float* attn    = c_t + B_ * H2;                       // [B,T]
  float* pg      = attn + B_ * T_;                      // [B,1]
  float* covnext = pg + B_;                             // [B,T]

  // workspace (~4 MB)
  float* w     = (float*)d_ws;
  float* Axc   = w;                    // B*KXC  = 81920
  float* x     = Axc + B_ * KXC;       // B*E    = 16384
  float* Ag    = x + B_ * E_;          // B*KG   = 49152
  float* Wg    = Ag + B_ * KG;         // H4*KG  = 393216
  float* gates = Wg + H4 * KG;         // B*H4   = 131072
  float* Adp   = gates + B_ * H4;      // B*H2   = 65536 (s_t_hat)
  float* dec   = Adp + B_ * H2;        // B*H2   = 65536
  float* scr   = dec + B_ * H2;        // B*T    = 51200
  float* Ao1   = scr + B_ * T_;        // B*K1K  = 98304
  float* out1  = Ao1 + B_ * K1K;       // B*H    = 32768

  // grid helper: blocks = ceil(ceil(N/16)/8 waves per block)
  #define GEMM_GRID(N) ((((N) + 15) / 16 + 7) / 8)

  k_cat_xc <<<(B_ * KXC + 255) / 256, 256, 0, stream>>>(y, ctm1, glove, Axc);
  k_gemm_wmma<0, KXC, E_><<<GEMM_GRID(E_), 256, 0, stream>>>(Axc, Wxc, bxc, x, E_);

  k_pack_wg<<<(H4 * KG + 255) / 256, 256, 0, stream>>>(Wih, Whh, Wg);
  k_cat_g  <<<(B_ * KG + 255) / 256, 256, 0, stream>>>(x, h0, Ag);
  k_gemm_wmma<0, KG, H4><<<GEMM_GRID(H4), 256, 0, stream>>>(Ag, Wg, (const float*)nullptr, gates, H4);

  k_lstm   <<<(B_ * H_ + 255) / 256, 256, 0, stream>>>(gates, c0, bih, bhh, h_new, c_new);

  k_cat_dp <<<(B_ * H2 + 255) / 256, 256, 0, stream>>>(h_new, c_new, Adp);
  k_gemm_wmma<0, H2, H2><<<GEMM_GRID(H2), 256, 0, stream>>>(Adp, Wdp, bdp, dec, H2);

  k_scores <<<B_ * (T_ / 8), 256, 0, stream>>>(ef, dec, cov, vatt, Wc, scr);
  k_attn   <<<B_, 512, 0, stream>>>(scr, mask, cov, attn, covnext);
  k_ct     <<<B_, 512, 0, stream>>>(attn, eo, c_t);
  k_pgen   <<<B_ / 4, 128, 0, stream>>>(c_t, Adp, x, Wpg, bpg, pg);

  k_cat_o1 <<<(B_ * K1K + 255) / 256, 256, 0, stream>>>(h_new, c_t, Ao1);
  k_gemm_wmma<0, K1K, H_><<<GEMM_GRID(H_), 256, 0, stream>>>(Ao1, W1, b1, out1, H_);

  // logits GEMM: A (128x256 = 128 KB) staged in dynamic LDS per block
  k_gemm_wmma<1, H_, VE><<<GEMM_GRID(V_), 256, B_ * H_ * sizeof(float), stream>>>(
      out1, W2, b2, fdist, V_);

  k_vsoftmax<<<B_, 1024, 0, stream>>>(pg, fdist);
  k_scatter <<<(B_ * T_ + 255) / 256, 256, 0, stream>>>(attn, pg, ext, fdist);

  #undef GEMM_GRID
}